// SelfAttention_84172769067025
// MI455X (gfx1250) — compile-verified
//
#include <hip/hip_runtime.h>
#include <hip/hip_bf16.h>

// Problem sizes (fixed by the reference).
constexpr int BB = 4;
constexpr int LL = 512;
constexpr int DD = 128;

typedef __attribute__((ext_vector_type(2))) float v2f;
typedef __attribute__((ext_vector_type(8))) float v8f;

// ---------- fast transcendentals (hw TRANS ops, guarded) ----------
__device__ __forceinline__ float fexp2(float x) {
#if __has_builtin(__builtin_amdgcn_exp2f)
  return __builtin_amdgcn_exp2f(x);
#else
  return exp2f(x);
#endif
}
__device__ __forceinline__ float frcp(float x) {
#if __has_builtin(__builtin_amdgcn_rcpf)
  return __builtin_amdgcn_rcpf(x);
#else
  return 1.0f / x;
#endif
}
__device__ __forceinline__ float ftanh(float x) {
#if __has_builtin(__builtin_amdgcn_tanhf)
  return __builtin_amdgcn_tanhf(x);   // single v_tanh_f32 (gfx1250 TRANS op)
#else
  // tanh(x) = 1 - 2/(exp2(2*log2e*x)+1); saturates correctly at +/-inf
  float e = fexp2(x * 2.8853900817779268f);
  return 1.0f - 2.0f * frcp(e + 1.0f);
#endif
}

// =================================================================
// Kernel 1: Q/K projection.  Q = X*Wq^T + bq, K = X*Wk^T + bk
// Grid: 256 blocks x 256 threads.
//   blockIdx.x[7]   -> which matrix (0=Q, 1=K)
//   blockIdx.x[6:0] -> 16-row tile of the flattened [B*L, D] input
//   wave (8/block)  -> 16-column tile of the output
// Each wave runs 32x V_WMMA_F32_16X16X4_F32 over K=128.
// =================================================================
__global__ __launch_bounds__(256) void qk_proj_kernel(
    const float* __restrict__ x,
    const float* __restrict__ Wq, const float* __restrict__ bq,
    const float* __restrict__ Wk, const float* __restrict__ bk,
    float* __restrict__ qout, float* __restrict__ kout)
{
  const int tid  = threadIdx.x;
  const int wave = tid >> 5;
  const int lane = tid & 31;
  const int which   = blockIdx.x >> 7;          // 0 = Q, 1 = K
  const int rowTile = blockIdx.x & 127;         // 128 tiles over B*L = 2048 rows

  const float* Wm   = which ? Wk : Wq;
  const float* bias = which ? bk : bq;
  float*       outp = which ? kout : qout;

  const int row0 = rowTile * 16;
  const int col0 = wave * 16;
  const int half = lane >> 4;                   // K half-wave split
  const int l15  = lane & 15;
  const int koff = 2 * half;                    // lanes 0-15: K=0,1  lanes 16-31: K=2,3

  v8f acc = {};
  for (int kk = 0; kk < DD / 4; ++kk) {
    const int kbase = 4 * kk + koff;
    // A frag: rows of X, two consecutive K elements -> one b64 load
    v2f a = *(const v2f*)&x[(row0 + l15) * DD + kbase];
    // B frag: B[k][n] = Wm[n][k]; two consecutive k in Wm's row -> one b64 load
    v2f bf = *(const v2f*)&Wm[(col0 + l15) * DD + kbase];
    acc = __builtin_amdgcn_wmma_f32_16x16x4_f32(
        false, a, false, bf, (short)0, acc, false, false);
  }

  const float bv = bias[col0 + l15];
  #pragma unroll
  for (int r = 0; r < 8; ++r) {
    const int m = r + 8 * half;                 // C/D layout: vgpr r, half-wave
    outp[(row0 + m) * DD + col0 + l15] = acc[r] + bv;
  }
}

// =================================================================
// Kernel 2: fused scores(tanh) + softmax + (P @ X) per (batch, 16-row tile).
// Grid: B*(L/16) = 128 blocks x 256 threads (8 wave32).
//   Phase 1: 16x512 score strip in LDS.  thread = (i=tid>>4, j=tid&15);
//            inner d-loop = 1 v_tanh + 1 fma per element, all LDS operands.
//   Phase 2: row softmax in LDS (exp2-based).
//   Phase 3: out[16x128] = P[16x512] @ x_b[512x128] via 128x WMMA f32 16x16x4
//            per wave (A = normalized P from LDS, B = x columns from L2).
// =================================================================
constexpr int STR_K = 132;   // k/q tile row stride (pad: 132 % 64 banks = 4)
constexpr int STR_S = 520;   // score strip row stride (even for b64, % 64 = 8)

__global__ __launch_bounds__(256) void attn_fused_kernel(
    const float* __restrict__ x,
    const float* __restrict__ q, const float* __restrict__ k,
    const float* __restrict__ w, const float* __restrict__ bw,
    float* __restrict__ out)
{
  __shared__ float qs[16 * STR_K];
  __shared__ float kt[16 * STR_K];
  __shared__ float wl[DD];
  __shared__ float sc[16 * STR_S];
  __shared__ float invl[16];

  const int tid   = threadIdx.x;
  const int b     = blockIdx.x >> 5;            // L/16 = 32 tiles per batch
  const int iTile = blockIdx.x & 31;
  const int i0    = iTile * 16;

  const float* qb = q + (size_t)b * LL * DD;
  const float* kb = k + (size_t)b * LL * DD;
  const float* xb = x + (size_t)b * LL * DD;

  // stage w and the q-tile
  if (tid < DD) wl[tid] = w[tid];
  for (int e = tid; e < 16 * DD; e += 256) {
    const int r = e >> 7, c = e & (DD - 1);
    qs[r * STR_K + c] = qb[(i0 + r) * DD + c];
  }
  __syncthreads();

  // ---------- Phase 1: scores ----------
  const float bwv = bw[0];
  const int si = tid >> 4;                      // query row within tile
  const int sj = tid & 15;                      // key row within j-chunk

  for (int j0 = 0; j0 < LL; j0 += 16) {
    for (int e = tid; e < 16 * DD; e += 256) {
      const int r = e >> 7, c = e & (DD - 1);
      kt[r * STR_K + c] = kb[(j0 + r) * DD + c];
    }
    __syncthreads();

    const float* qrow = &qs[si * STR_K];
    const float* krow = &kt[sj * STR_K];
    float s = 0.f;
    #pragma unroll 8
    for (int d = 0; d < DD; ++d)
      s += wl[d] * ftanh(qrow[d] + krow[d]);    // TRANS + co-executing FMA
    sc[si * STR_S + (j0 + sj)] = s + bwv;
    __syncthreads();                            // kt consumed before reload
  }

  // ---------- Phase 2: softmax ----------
  if (tid < 16) {
    float* row = &sc[tid * STR_S];
    float m = row[0];
    for (int j = 1; j < LL; ++j) m = fmaxf(m, row[j]);
    float sum = 0.f;
    for (int j = 0; j < LL; ++j) {
      const float p = fexp2((row[j] - m) * 1.4426950408889634f);
      row[j] = p;
      sum += p;
    }
    invl[tid] = frcp(sum);
  }
  __syncthreads();
  for (int e = tid; e < 16 * LL; e += 256) {    // normalize P in parallel
    const int r = e >> 9, c = e & (LL - 1);
    sc[r * STR_S + c] *= invl[r];
  }
  __syncthreads();

  // ---------- Phase 3: out = P @ x_b ----------
  const int wave = tid >> 5;
  const int lane = tid & 31;
  const int half = lane >> 4;
  const int l15  = lane & 15;
  const int koff = 2 * half;
  const int n0   = wave * 16;                   // wave owns a 16-wide d-slice

  v8f acc = {};
  for (int kk = 0; kk < LL / 4; ++kk) {
    const int kbase = 4 * kk + koff;
    // A frag from normalized P in LDS (b64, 8B aligned: STR_S even, kbase even)
    v2f a = *(const v2f*)&sc[l15 * STR_S + kbase];
    // B frag: B[k][n] = x_b[j=k][n] (column gather, stride D)
    v2f bf;
    bf[0] = xb[(kbase + 0) * DD + n0 + l15];
    bf[1] = xb[(kbase + 1) * DD + n0 + l15];
    acc = __builtin_amdgcn_wmma_f32_16x16x4_f32(
        false, a, false, bf, (short)0, acc, false, false);
  }

  float* ob = out + (size_t)b * LL * DD;
  #pragma unroll
  for (int r = 0; r < 8; ++r) {
    const int m = r + 8 * half;
    ob[(i0 + m) * DD + n0 + l15] = acc[r];
  }
}

// =================================================================
extern "C" void kernel_launch(void* const* d_in, const int* in_sizes, int n_in,
                              void* d_out, int out_size, void* d_ws, size_t ws_size,
                              hipStream_t stream) {
  (void)in_sizes; (void)n_in; (void)out_size; (void)ws_size;
  const float* x  = (const float*)d_in[0];
  // d_in[1] = mask: unused — the reference's masked_fill is out-of-place and
  // discarded, so the mask has no effect on the output.
  const float* Wq = (const float*)d_in[2];
  const float* bq = (const float*)d_in[3];
  const float* Wk = (const float*)d_in[4];
  const float* bk = (const float*)d_in[5];
  const float* w  = (const float*)d_in[6];
  const float* bw = (const float*)d_in[7];
  float* outp = (float*)d_out;

  float* qbuf = (float*)d_ws;                   // [B*L*D] = 1 MB
  float* kbuf = qbuf + (size_t)BB * LL * DD;    // [B*L*D] = 1 MB

  qk_proj_kernel<<<256, 256, 0, stream>>>(x, Wq, bq, Wk, bk, qbuf, kbuf);
  attn_fused_kernel<<<BB * (LL / 16), 256, 0, stream>>>(x, qbuf, kbuf, w, bw, outp);
}